// GCN_78554951844320
// MI455X (gfx1250) — compile-verified
//
#include <hip/hip_runtime.h>

typedef __attribute__((ext_vector_type(2))) float v2f;
typedef __attribute__((ext_vector_type(8))) float v8f;

// ---------------------------------------------------------------------------
// Utility: fill a float buffer
// ---------------------------------------------------------------------------
__global__ __launch_bounds__(256) void fill_kernel(float* __restrict__ p, long n, float v) {
    long i = (long)blockIdx.x * blockDim.x + threadIdx.x;
    if (i < n) p[i] = v;
}

// ---------------------------------------------------------------------------
// Degree histogram: deg[row[e]] += 1  (float atomics, lands in L2)
// ---------------------------------------------------------------------------
__global__ __launch_bounds__(256) void deg_kernel(const int* __restrict__ row,
                                                  float* __restrict__ deg, int E) {
    int e = blockIdx.x * blockDim.x + threadIdx.x;
    if (e < E) atomicAdd(deg + row[e], 1.0f);
}

// dinv[i] = rsqrt(deg_edges[i] + 1)   (+1 = self loop; always > 0)
__global__ __launch_bounds__(256) void dinv_kernel(float* __restrict__ deg, int n) {
    int i = blockIdx.x * blockDim.x + threadIdx.x;
    if (i < n) deg[i] = rsqrtf(deg[i] + 1.0f);
}

// ---------------------------------------------------------------------------
// FP32 WMMA GEMM: OUT[M,N] = X[M,K] @ W[K,N]
// One wave computes one 16x16 C tile via V_WMMA_F32_16X16X4_F32.
// Block = 128 threads (4 waves); grid = (M/16, N/64); wave w owns n-tile
// blockIdx.y*4 + w.
//
// Per ISA 7.12.2 (32-bit layouts, wave32):
//   A 16x4 : lane l<16 holds row M=l, VGPR0=K+0, VGPR1=K+1;
//            lane l>=16 holds row M=l-16, VGPR0=K+2, VGPR1=K+3.
//   B 4x16 : VGPR0 lanes0-15 = row K+0, lanes16-31 = row K+2;
//            VGPR1 lanes0-15 = row K+1, lanes16-31 = row K+3. (cols = lane&15)
//   C/D 16x16: VGPR g = row g (lanes 0-15) / row g+8 (lanes 16-31).
// ---------------------------------------------------------------------------
__global__ __launch_bounds__(128)
void gemm_wmma_f32(const float* __restrict__ X, const float* __restrict__ W,
                   float* __restrict__ OUT, int M, int K, int N) {
    const int wave = threadIdx.x >> 5;
    const int lane = threadIdx.x & 31;
    const int m0 = blockIdx.x << 4;
    const int n0 = ((blockIdx.y << 2) + wave) << 4;
    if (m0 >= M || n0 >= N) return;

    const int half = lane >> 4;   // 0: lanes 0-15, 1: lanes 16-31
    const int l    = lane & 15;

    const float* xrow = X + (long)(m0 + l) * K + (half << 1);   // A row, K base
    const float* wcol = W + (long)(half << 1) * N + n0 + l;     // B col, K base

    v8f acc = {};
    for (int k = 0; k < K; k += 4) {
        v2f a, b;
        a.x = xrow[k];
        a.y = xrow[k + 1];
        b.x = wcol[(long)k * N];
        b.y = wcol[(long)(k + 1) * N];
        acc = __builtin_amdgcn_wmma_f32_16x16x4_f32(
            /*neg_a=*/false, a, /*neg_b=*/false, b,
            /*c_mod=*/(short)0, acc, /*reuse_a=*/false, /*reuse_b=*/false);
    }

    float* orow = OUT + (long)(m0 + (half << 3)) * N + n0 + l;
#pragma unroll
    for (int g = 0; g < 8; ++g)
        orow[(long)g * N] = acc[g];
}

// ---------------------------------------------------------------------------
// Edge aggregation: agg[row[e]] += dinv[row[e]]*dinv[col[e]] * h[col[e]]
// One wave per edge; each lane handles one float4 chunk of the feature row
// (32 lanes x 4 floats = up to 128 features). Gather is a single
// global_load_b128 per lane; scatter is global_atomic_add_f32 (hits L2).
// ---------------------------------------------------------------------------
__global__ __launch_bounds__(256)
void agg_kernel(const float* __restrict__ h, const int* __restrict__ row,
                const int* __restrict__ col, const float* __restrict__ dinv,
                float* __restrict__ agg, int E, int F) {
    int gw = blockIdx.x * (blockDim.x >> 5) + (threadIdx.x >> 5);
    if (gw >= E) return;
    const int lane = threadIdx.x & 31;

    const int r = row[gw];
    const int c = col[gw];
    const float w = dinv[r] * dinv[c];

    const int f = lane << 2;
    if (f < F) {
        const float4 v = *(const float4*)(h + (long)c * F + f);
        float* dst = agg + (long)r * F + f;
        atomicAdd(dst + 0, w * v.x);
        atomicAdd(dst + 1, w * v.y);
        atomicAdd(dst + 2, w * v.z);
        atomicAdd(dst + 3, w * v.w);
    }
}

// ---------------------------------------------------------------------------
// Epilogue: out = (relu?)(agg + dinv[node]^2 * h + bias)
// Folds the self-loop message in stream order (no atomics needed).
// ---------------------------------------------------------------------------
__global__ __launch_bounds__(256)
void epi_kernel(const float* __restrict__ agg, const float* __restrict__ h,
                const float* __restrict__ dinv, const float* __restrict__ bias,
                float* __restrict__ out, long total, int fshift, int fmask, int doRelu) {
    long i = (long)blockIdx.x * blockDim.x + threadIdx.x;
    if (i >= total) return;
    const int node = (int)(i >> fshift);
    const int f    = (int)i & fmask;
    const float s  = dinv[node];
    float v = agg[i] + s * s * h[i] + bias[f];
    out[i] = doRelu ? fmaxf(v, 0.0f) : v;
}

// ---------------------------------------------------------------------------
// Launch: 3-layer GCN
// ws layout (floats): [dinv: N][B1: N*128][B2: N*128]  (~51.4 MB)
// ---------------------------------------------------------------------------
extern "C" void kernel_launch(void* const* d_in, const int* in_sizes, int n_in,
                              void* d_out, int out_size, void* d_ws, size_t ws_size,
                              hipStream_t stream) {
    const float* feat = (const float*)d_in[0];
    const float* W1   = (const float*)d_in[1];
    const float* b1   = (const float*)d_in[2];
    const float* W2   = (const float*)d_in[3];
    const float* b2   = (const float*)d_in[4];
    const float* W3   = (const float*)d_in[5];
    const float* b3   = (const float*)d_in[6];
    const int*   ei   = (const int*)d_in[7];

    const int F_in = 64, H = 128, F_out = 64;
    const int Nn = in_sizes[0] / F_in;   // 50000
    const int E  = in_sizes[7] / 2;      // 800000
    const int* row = ei;                 // edge_index[0]
    const int* col = ei + E;             // edge_index[1]

    float* dinv = (float*)d_ws;
    float* B1   = dinv + Nn;             // N*128 (offsets stay 16B-aligned)
    float* B2   = B1 + (long)Nn * H;
    float* out  = (float*)d_out;

    const dim3 blk256(256);
    const int mT = (Nn + 15) / 16;
    const long nH  = (long)Nn * H;
    const long nFo = (long)Nn * F_out;
    const int edgeBlocks = (E + 7) / 8;  // 8 waves per 256-thread block

    // --- symmetric normalization: dinv = rsqrt(deg + 1) ---------------------
    fill_kernel<<<(Nn + 255) / 256, blk256, 0, stream>>>(dinv, Nn, 0.0f);
    deg_kernel<<<(E + 255) / 256, blk256, 0, stream>>>(row, dinv, E);
    dinv_kernel<<<(Nn + 255) / 256, blk256, 0, stream>>>(dinv, Nn);

    // --- layer 1: x1 = relu(Agg(feat @ W1) + b1) ----------------------------
    gemm_wmma_f32<<<dim3(mT, H / 64), dim3(128), 0, stream>>>(feat, W1, B1, Nn, F_in, H);
    fill_kernel<<<(nH + 255) / 256, blk256, 0, stream>>>(B2, nH, 0.0f);
    agg_kernel<<<edgeBlocks, blk256, 0, stream>>>(B1, row, col, dinv, B2, E, H);
    epi_kernel<<<(nH + 255) / 256, blk256, 0, stream>>>(B2, B1, dinv, b1, B1, nH, 7, 127, 1);

    // --- layer 2: x2 = relu(Agg(x1 @ W2) + b2) ------------------------------
    gemm_wmma_f32<<<dim3(mT, H / 64), dim3(128), 0, stream>>>(B1, W2, B2, Nn, H, H);
    fill_kernel<<<(nH + 255) / 256, blk256, 0, stream>>>(B1, nH, 0.0f);
    agg_kernel<<<edgeBlocks, blk256, 0, stream>>>(B2, row, col, dinv, B1, E, H);
    epi_kernel<<<(nH + 255) / 256, blk256, 0, stream>>>(B1, B2, dinv, b2, B2, nH, 7, 127, 1);

    // --- layer 3: out = Agg(x2 @ W3) + b3  (no relu) ------------------------
    gemm_wmma_f32<<<dim3(mT, F_out / 64), dim3(128), 0, stream>>>(B2, W3, B1, Nn, H, F_out);
    fill_kernel<<<(nFo + 255) / 256, blk256, 0, stream>>>(out, nFo, 0.0f);
    agg_kernel<<<edgeBlocks, blk256, 0, stream>>>(B1, row, col, dinv, out, E, F_out);
    epi_kernel<<<(nFo + 255) / 256, blk256, 0, stream>>>(out, B1, dinv, b3, out, nFo, 6, 63, 0);
}